// FlashRetentionBody_77988016160779
// MI455X (gfx1250) — compile-verified
//
#include <hip/hip_runtime.h>
#include <stdint.h>

// FlashRetentionBody: qkm = qk*m; r_new = max(r_wo_clamp + rowsum(|qkm|), 1);
// acco_out = acco * r / r_new (j>0); acc = qkm / r_new.
// Bandwidth-bound (671 MB @ 23.3 TB/s ~ 29 us). One block per row, qkm held in
// VGPRs between passes, acco streamed global->LDS with CDNA5 async DMA loads.

typedef float f4 __attribute__((ext_vector_type(4)));

#define COLS 8192
#define TPB  256
#define F4PT (COLS / 4 / TPB)   // 8 float4 chunks per thread

__global__ __launch_bounds__(TPB) void fr_body_kernel(
    const float* __restrict__ qk,
    const float* __restrict__ mm,
    const float* __restrict__ acco,
    const float* __restrict__ r,
    const float* __restrict__ rwc,
    const int*  __restrict__ jptr,
    float* __restrict__ out_acco,
    float* __restrict__ out_acc,
    float* __restrict__ out_rnew)
{
    __shared__ f4    s_acco[COLS / 4];      // 32 KB: acco row staged by async DMA
    __shared__ float s_wsum[TPB / 32];      // per-wave partial sums
    __shared__ float s_bc[3];               // r_new, scale, 1/r_new broadcast

    const int    t    = threadIdx.x;
    const int    row  = blockIdx.x;
    const size_t base = (size_t)row * COLS;

    // ---- issue async global->LDS staging of the acco row (overlaps pass 1) ----
    // CDNA5 GLOBAL_LOAD_ASYNC_TO_LDS_B128, GVS mode: saddr = uniform row base,
    // vaddr = per-lane byte offset, vdst = LDS byte address. Tracked by ASYNCcnt.
    {
        unsigned long long gbase = (unsigned long long)(uintptr_t)(acco + base);
        unsigned int lds0 = (unsigned int)(uintptr_t)(void*)&s_acco[0];
        #pragma unroll
        for (int i = 0; i < F4PT; ++i) {
            unsigned int boff  = (unsigned int)((i * TPB + t) * 16);
            unsigned int laddr = lds0 + boff;
            asm volatile("global_load_async_to_lds_b128 %0, %1, %2"
                         :: "v"(laddr), "v"(boff), "s"(gbase) : "memory");
        }
    }

    // ---- pass 1: qkm = qk*m (kept in VGPRs), partial = sum |qkm| ----
    const f4* gqk = (const f4*)(qk + base);
    const f4* gm  = (const f4*)(mm + base);

    f4    qkm[F4PT];
    float part = 0.0f;
    #pragma unroll
    for (int i = 0; i < F4PT; ++i) {
        const int q = i * TPB + t;                  // lane-contiguous B128
        f4 a = __builtin_nontemporal_load(&gqk[q]); // streaming: th=NT
        f4 b = __builtin_nontemporal_load(&gm[q]);
        f4 p = a * b;
        qkm[i] = p;
        part += __builtin_fabsf(p.x) + __builtin_fabsf(p.y)
              + __builtin_fabsf(p.z) + __builtin_fabsf(p.w);
    }

    // ---- row reduction: wave32 shuffle tree + LDS across 8 waves ----
    #pragma unroll
    for (int off = 16; off > 0; off >>= 1)
        part += __shfl_xor(part, off, 32);
    if ((t & 31) == 0) s_wsum[t >> 5] = part;
    __syncthreads();

    if (t == 0) {
        float tot = 0.0f;
        #pragma unroll
        for (int w = 0; w < TPB / 32; ++w) tot += s_wsum[w];
        float rn = fmaxf(rwc[row] + tot, 1.0f);
        s_bc[0] = rn;
        s_bc[1] = (jptr[0] > 0) ? (r[row] / rn) : 1.0f;  // acco rescale factor
        s_bc[2] = 1.0f / rn;                             // acc normalizer
        out_rnew[row] = rn;                              // (M,1) output
    }
    __syncthreads();

    const float scale = s_bc[1];
    const float inv   = s_bc[2];

    // acco row must be resident in LDS before pass 2 consumes it
    asm volatile("s_wait_asynccnt 0x0" ::: "memory");

    // ---- pass 2: write acc = qkm/r_new and acco_out = acco * r/r_new ----
    f4* oacco = (f4*)(out_acco + base);
    f4* oacc  = (f4*)(out_acc  + base);
    #pragma unroll
    for (int i = 0; i < F4PT; ++i) {
        const int q = i * TPB + t;
        f4 accv = qkm[i] * inv;
        __builtin_nontemporal_store(accv, &oacc[q]);
        f4 aov = s_acco[q] * scale;                  // ds_load_b128 from staged row
        __builtin_nontemporal_store(aov, &oacco[q]);
    }
}

extern "C" void kernel_launch(void* const* d_in, const int* in_sizes, int n_in,
                              void* d_out, int out_size, void* d_ws, size_t ws_size,
                              hipStream_t stream) {
    const float* qk   = (const float*)d_in[0];
    const float* m    = (const float*)d_in[1];
    const float* acco = (const float*)d_in[2];
    const float* r    = (const float*)d_in[3];
    const float* rwc  = (const float*)d_in[4];
    const int*   j    = (const int*)d_in[5];

    const int    rows = in_sizes[3];          // M, from r's (M,1) shape
    const size_t mn   = (size_t)rows * COLS;

    float* out      = (float*)d_out;
    float* out_acco = out;                    // tuple order: acco_out, acc, r_new
    float* out_acc  = out + mn;
    float* out_rnew = out + 2 * mn;

    fr_body_kernel<<<dim3(rows), dim3(TPB), 0, stream>>>(
        qk, m, acco, r, rwc, j, out_acco, out_acc, out_rnew);
}